// Mamba1ICL_52295521796380
// MI455X (gfx1250) — compile-verified
//
#include <hip/hip_runtime.h>
#include <math.h>

// ---------------- problem constants ----------------
#define D_MODEL 256
#define D_STATE 16
#define D_CONV  4
#define D_INNER 512
#define DT_RANK 16
#define BATCH   4
#define NPTS    1024
#define SEQLEN  (2 * NPTS)            // 2048
#define MROWS   (BATCH * SEQLEN)      // 8192 total rows
#define NXZ     (2 * D_INNER)         // 1024
#define NDBL    (DT_RANK + 2 * D_STATE) // 48
#define NCHUNK  16
#define CHUNKT  (SEQLEN / NCHUNK)     // 128

typedef float v2f __attribute__((ext_vector_type(2)));
typedef float v8f __attribute__((ext_vector_type(8)));

__device__ __forceinline__ float siluf(float x) { return x / (1.f + __expf(-x)); }
__device__ __forceinline__ float softplusf(float x) {
  return (x > 20.f) ? x : log1pf(__expf(x));
}

// fp32 WMMA 16x16x4: D = A(16x4) * B(4x16) + C, exact fp32 MACs.
__device__ __forceinline__ v8f wmma4(v2f a, v2f b, v8f c) {
  return __builtin_amdgcn_wmma_f32_16x16x4_f32(false, a, false, b, (short)0, c,
                                               false, false);
}

// ---------------- 1) interleave x / projected y into seq [B,L,256] -------------
__global__ void k_build_seq(const float* __restrict__ x, const float* __restrict__ y,
                            const float* __restrict__ yproj_w,
                            const float* __restrict__ yproj_b,
                            float* __restrict__ seq) {
  int idx = blockIdx.x * blockDim.x + threadIdx.x;
  if (idx >= BATCH * SEQLEN * D_MODEL) return;
  int m  = idx % D_MODEL;
  int bt = idx / D_MODEL;
  int t  = bt % SEQLEN;
  int b  = bt / SEQLEN;
  int l  = t >> 1;
  float v;
  if ((t & 1) == 0) v = x[(size_t)(b * NPTS + l) * D_MODEL + m];
  else              v = y[b * NPTS + l] * yproj_w[m] + yproj_b[m];
  seq[idx] = v;
}

// ---------------- 2) xz = seq @ in_proj_w^T  (8192x256 @ 256x1024) -------------
// one wave -> 16x64 output strip (4 accumulators), K-step = 4 (fp32 WMMA)
__global__ __launch_bounds__(128) void k_gemm_inproj(const float* __restrict__ seq,
                                                     const float* __restrict__ w,
                                                     float* __restrict__ xz) {
  const int lane  = threadIdx.x & 31;
  const int warp  = threadIdx.x >> 5;
  const int wave  = blockIdx.x * 4 + warp;     // 0..8191
  const int ngrp  = wave & 15;                 // 16 groups of 64 cols
  const int mtile = wave >> 4;                 // 0..511
  const int hl    = lane >> 4;                 // which K-pair half
  const int r16   = lane & 15;                 // A: M-row / B,C: N-col
  const int mb    = mtile * 16;
  const int nb    = ngrp * 64;

  const float* arow = seq + (size_t)(mb + r16) * D_MODEL + 2 * hl;
  const float* br0  = w + (size_t)(nb +  0 + r16) * D_MODEL + 2 * hl;
  const float* br1  = w + (size_t)(nb + 16 + r16) * D_MODEL + 2 * hl;
  const float* br2  = w + (size_t)(nb + 32 + r16) * D_MODEL + 2 * hl;
  const float* br3  = w + (size_t)(nb + 48 + r16) * D_MODEL + 2 * hl;

  v8f c0 = {}, c1 = {}, c2 = {}, c3 = {};
#pragma unroll 4
  for (int k0 = 0; k0 < D_MODEL; k0 += 4) {
    __builtin_prefetch(arow + k0 + 32, 0, 0);      // global_prefetch_b8
    v2f a  = *(const v2f*)(arow + k0);
    v2f b0 = *(const v2f*)(br0 + k0);
    v2f b1 = *(const v2f*)(br1 + k0);
    v2f b2 = *(const v2f*)(br2 + k0);
    v2f b3 = *(const v2f*)(br3 + k0);
    c0 = wmma4(a, b0, c0);
    c1 = wmma4(a, b1, c1);
    c2 = wmma4(a, b2, c2);
    c3 = wmma4(a, b3, c3);
  }
#pragma unroll
  for (int r = 0; r < 8; ++r) {
    size_t o = (size_t)(mb + r + 8 * hl) * NXZ + nb + r16;
    xz[o]      = c0[r];
    xz[o + 16] = c1[r];
    xz[o + 32] = c2[r];
    xz[o + 48] = c3[r];
  }
}

// ---------------- 3) causal depthwise conv (4 taps) + SiLU ----------------
__global__ void k_conv_silu(const float* __restrict__ xz,
                            const float* __restrict__ conv_w,
                            const float* __restrict__ conv_b,
                            float* __restrict__ xc) {
  int idx = blockIdx.x * blockDim.x + threadIdx.x;
  if (idx >= MROWS * D_INNER) return;
  int d  = idx % D_INNER;
  int bt = idx / D_INNER;
  int t  = bt % SEQLEN;
  int b  = bt / SEQLEN;
  float acc = conv_b[d];
#pragma unroll
  for (int k = 0; k < D_CONV; ++k) {
    int tt = t - (D_CONV - 1) + k;
    if (tt >= 0)
      acc += xz[(size_t)(b * SEQLEN + tt) * NXZ + d] * conv_w[d * D_CONV + k];
  }
  xc[idx] = siluf(acc);
}

// keep z at the final timestep (only place z is ever used)
__global__ void k_zlast(const float* __restrict__ xz, float* __restrict__ zlast) {
  int idx = blockIdx.x * blockDim.x + threadIdx.x;
  if (idx >= BATCH * D_INNER) return;
  int d = idx % D_INNER, b = idx / D_INNER;
  zlast[idx] = xz[(size_t)(b * SEQLEN + SEQLEN - 1) * NXZ + D_INNER + d];
}

// ---------------- 4) dbl = xc @ x_proj_w^T  (8192x512 @ 512x48) ----------------
__global__ __launch_bounds__(128) void k_gemm_xproj(const float* __restrict__ xc,
                                                    const float* __restrict__ w,
                                                    float* __restrict__ dbl) {
  const int lane  = threadIdx.x & 31;
  const int warp  = threadIdx.x >> 5;
  const int mtile = blockIdx.x * 4 + warp;     // 0..511
  const int hl    = lane >> 4;
  const int r16   = lane & 15;
  const int mb    = mtile * 16;

  const float* arow = xc + (size_t)(mb + r16) * D_INNER + 2 * hl;
  const float* br0  = w + (size_t)( 0 + r16) * D_INNER + 2 * hl;
  const float* br1  = w + (size_t)(16 + r16) * D_INNER + 2 * hl;
  const float* br2  = w + (size_t)(32 + r16) * D_INNER + 2 * hl;

  v8f c0 = {}, c1 = {}, c2 = {};
#pragma unroll 4
  for (int k0 = 0; k0 < D_INNER; k0 += 4) {
    v2f a  = *(const v2f*)(arow + k0);
    v2f b0 = *(const v2f*)(br0 + k0);
    v2f b1 = *(const v2f*)(br1 + k0);
    v2f b2 = *(const v2f*)(br2 + k0);
    c0 = wmma4(a, b0, c0);
    c1 = wmma4(a, b1, c1);
    c2 = wmma4(a, b2, c2);
  }
#pragma unroll
  for (int r = 0; r < 8; ++r) {
    size_t o = (size_t)(mb + r + 8 * hl) * NDBL + r16;
    dbl[o]      = c0[r];
    dbl[o + 16] = c1[r];
    dbl[o + 32] = c2[r];
  }
}

// ---------------- 5) delta = softplus(dt @ dt_proj_w^T + b)  (K=16) ------------
__global__ __launch_bounds__(128) void k_gemm_dtproj(const float* __restrict__ dbl,
                                                     const float* __restrict__ w,
                                                     const float* __restrict__ bias,
                                                     float* __restrict__ delta) {
  const int lane  = threadIdx.x & 31;
  const int warp  = threadIdx.x >> 5;
  const int wave  = blockIdx.x * 4 + warp;     // 0..4095
  const int ngrp  = wave & 7;                  // 8 groups of 64 cols
  const int mtile = wave >> 3;                 // 0..511
  const int hl    = lane >> 4;
  const int r16   = lane & 15;
  const int mb    = mtile * 16;
  const int nb    = ngrp * 64;

  const float* arow = dbl + (size_t)(mb + r16) * NDBL + 2 * hl;  // dt = cols 0..15
  const float* br0  = w + (size_t)(nb +  0 + r16) * DT_RANK + 2 * hl;
  const float* br1  = w + (size_t)(nb + 16 + r16) * DT_RANK + 2 * hl;
  const float* br2  = w + (size_t)(nb + 32 + r16) * DT_RANK + 2 * hl;
  const float* br3  = w + (size_t)(nb + 48 + r16) * DT_RANK + 2 * hl;

  v8f c0 = {}, c1 = {}, c2 = {}, c3 = {};
#pragma unroll
  for (int k0 = 0; k0 < DT_RANK; k0 += 4) {
    v2f a  = *(const v2f*)(arow + k0);
    v2f b0 = *(const v2f*)(br0 + k0);
    v2f b1 = *(const v2f*)(br1 + k0);
    v2f b2 = *(const v2f*)(br2 + k0);
    v2f b3 = *(const v2f*)(br3 + k0);
    c0 = wmma4(a, b0, c0);
    c1 = wmma4(a, b1, c1);
    c2 = wmma4(a, b2, c2);
    c3 = wmma4(a, b3, c3);
  }
#pragma unroll
  for (int r = 0; r < 8; ++r) {
    size_t mrow = (size_t)(mb + r + 8 * hl);
    int n0 = nb + r16;
    delta[mrow * D_INNER + n0]      = softplusf(c0[r] + bias[n0]);
    delta[mrow * D_INNER + n0 + 16] = softplusf(c1[r] + bias[n0 + 16]);
    delta[mrow * D_INNER + n0 + 32] = softplusf(c2[r] + bias[n0 + 32]);
    delta[mrow * D_INNER + n0 + 48] = softplusf(c3[r] + bias[n0 + 48]);
  }
}

// ---------------- 6) chunked exclusive suffix sum of delta over t --------------
__global__ __launch_bounds__(512) void k_sufpart(const float* __restrict__ delta,
                                                 float* __restrict__ P) {
  int b = blockIdx.x >> 4, c = blockIdx.x & 15, d = threadIdx.x;
  float s = 0.f;
  for (int i = 0; i < CHUNKT; ++i)
    s += delta[(size_t)(b * SEQLEN + c * CHUNKT + i) * D_INNER + d];
  P[(size_t)(b * NCHUNK + c) * D_INNER + d] = s;
}

__global__ __launch_bounds__(512) void k_sufwrite(const float* __restrict__ delta,
                                                  const float* __restrict__ P,
                                                  float* __restrict__ S) {
  int b = blockIdx.x >> 4, c = blockIdx.x & 15, d = threadIdx.x;
  float run = 0.f;
  for (int c2 = c + 1; c2 < NCHUNK; ++c2)
    run += P[(size_t)(b * NCHUNK + c2) * D_INNER + d];
  for (int i = CHUNKT - 1; i >= 0; --i) {
    size_t o = (size_t)(b * SEQLEN + c * CHUNKT + i) * D_INNER + d;
    S[o] = run;            // exclusive: sum over s > t
    run += delta[o];
  }
}

// ---------------- 7) h_final-as-reduction:  y_d += Δx · Σ_n B_t C_L e^{A S} ----
__global__ __launch_bounds__(512) void k_scan_reduce(const float* __restrict__ delta,
                                                     const float* __restrict__ xc,
                                                     const float* __restrict__ S,
                                                     const float* __restrict__ dbl,
                                                     const float* __restrict__ A_log,
                                                     float* __restrict__ ypart) {
  int b = blockIdx.x >> 4, c = blockIdx.x & 15, d = threadIdx.x;
  __shared__ float sC[D_STATE];
  __shared__ float sBC[D_STATE];
  if (d < D_STATE)
    sC[d] = dbl[(size_t)(b * SEQLEN + SEQLEN - 1) * NDBL + DT_RANK + D_STATE + d];
  float An[D_STATE];
#pragma unroll
  for (int n = 0; n < D_STATE; ++n) An[n] = -__expf(A_log[d * D_STATE + n]);
  __syncthreads();

  float acc = 0.f;
  for (int i = 0; i < CHUNKT; ++i) {
    int t = c * CHUNKT + i;
    __syncthreads();
    if (d < D_STATE)
      sBC[d] = dbl[(size_t)(b * SEQLEN + t) * NDBL + DT_RANK + d] * sC[d];
    __syncthreads();
    size_t o = (size_t)(b * SEQLEN + t) * D_INNER + d;
    float g = delta[o] * xc[o];
    float s = S[o];
    float wsum = 0.f;
#pragma unroll
    for (int n = 0; n < D_STATE; ++n) wsum += sBC[n] * __expf(An[n] * s);
    acc += g * wsum;
  }
  ypart[(size_t)(b * NCHUNK + c) * D_INNER + d] = acc;
}

// ---------------- 8) finalize: gate with silu(z), fold out_proj∘head -----------
__global__ __launch_bounds__(512) void k_finalize(const float* __restrict__ ypart,
                                                  const float* __restrict__ xc,
                                                  const float* __restrict__ zlast,
                                                  const float* __restrict__ Dvec,
                                                  const float* __restrict__ out_proj_w,
                                                  const float* __restrict__ head_w,
                                                  const float* __restrict__ head_b,
                                                  float* __restrict__ out) {
  int b = blockIdx.x, d = threadIdx.x;
  float y = 0.f;
  for (int c = 0; c < NCHUNK; ++c)
    y += ypart[(size_t)(b * NCHUNK + c) * D_INNER + d];
  y += xc[(size_t)(b * SEQLEN + SEQLEN - 1) * D_INNER + d] * Dvec[d];
  y *= siluf(zlast[b * D_INNER + d]);
  float wc = 0.f;
  for (int m = 0; m < D_MODEL; ++m)
    wc += head_w[m] * out_proj_w[(size_t)m * D_INNER + d];
  __shared__ float red[D_INNER];
  red[d] = y * wc;
  __syncthreads();
  for (int s = D_INNER / 2; s > 0; s >>= 1) {
    if (d < s) red[d] += red[d + s];
    __syncthreads();
  }
  if (d == 0) out[b] = red[0] + head_b[0];
}

// ---------------- driver ----------------
extern "C" void kernel_launch(void* const* d_in, const int* in_sizes, int n_in,
                              void* d_out, int out_size, void* d_ws, size_t ws_size,
                              hipStream_t stream) {
  (void)in_sizes; (void)n_in; (void)out_size; (void)ws_size;
  const float* x          = (const float*)d_in[0];
  const float* y          = (const float*)d_in[1];
  const float* in_proj_w  = (const float*)d_in[2];
  const float* conv_w     = (const float*)d_in[3];
  const float* conv_b     = (const float*)d_in[4];
  const float* x_proj_w   = (const float*)d_in[5];
  const float* dt_proj_w  = (const float*)d_in[6];
  const float* dt_proj_b  = (const float*)d_in[7];
  const float* A_log      = (const float*)d_in[8];
  const float* Dvec       = (const float*)d_in[9];
  const float* out_proj_w = (const float*)d_in[10];
  const float* yproj_w    = (const float*)d_in[11];
  const float* yproj_b    = (const float*)d_in[12];
  const float* head_w     = (const float*)d_in[13];
  const float* head_b     = (const float*)d_in[14];

  // workspace layout (floats); dead buffers are aliased -> ~56 MiB total
  float* ws    = (float*)d_ws;
  float* seq   = ws + 0;                 // [8192,256]  (dead after in_proj GEMM)
  float* dbl   = ws + 0;                 // [8192,48]   aliases seq
  float* zlast = ws + 393216;            // [4,512]
  float* P     = ws + 395264;            // [4,16,512]
  float* ypart = ws + 428032;            // [4,16,512]
  float* xz    = ws + 2097152;           // [8192,1024] (dead after conv+zlast)
  float* delta = ws + 2097152;           // [8192,512]  aliases lower xz
  float* Ssuf  = ws + 6291456;           // [8192,512]  aliases upper xz
  float* xc    = ws + 10485760;          // [8192,512]

  k_build_seq<<<(BATCH * SEQLEN * D_MODEL + 255) / 256, 256, 0, stream>>>(
      x, y, yproj_w, yproj_b, seq);
  k_gemm_inproj<<<(512 * 16) / 4, 128, 0, stream>>>(seq, in_proj_w, xz);
  k_conv_silu<<<(MROWS * D_INNER + 255) / 256, 256, 0, stream>>>(
      xz, conv_w, conv_b, xc);
  k_zlast<<<(BATCH * D_INNER + 255) / 256, 256, 0, stream>>>(xz, zlast);
  k_gemm_xproj<<<512 / 4, 128, 0, stream>>>(xc, x_proj_w, dbl);
  k_gemm_dtproj<<<(512 * 8) / 4, 128, 0, stream>>>(dbl, dt_proj_w, dt_proj_b,
                                                   delta);
  k_sufpart<<<BATCH * NCHUNK, D_INNER, 0, stream>>>(delta, P);
  k_sufwrite<<<BATCH * NCHUNK, D_INNER, 0, stream>>>(delta, P, Ssuf);
  k_scan_reduce<<<BATCH * NCHUNK, D_INNER, 0, stream>>>(delta, xc, Ssuf, dbl,
                                                        A_log, ypart);
  k_finalize<<<BATCH, D_INNER, 0, stream>>>(ypart, xc, zlast, Dvec, out_proj_w,
                                            head_w, head_b, (float*)d_out);
}